// PointWiseConvolutionX_88175678587635
// MI455X (gfx1250) — compile-verified
//
#include <hip/hip_runtime.h>

#define N_PTS 8192
#define TILE 16
#define WAVES_PER_BLOCK 16
#define THREADS (WAVES_PER_BLOCK * 32)                 // 512
#define NBLOCKS (N_PTS / (TILE * WAVES_PER_BLOCK))     // 32

typedef float v2f __attribute__((ext_vector_type(2)));
typedef float v4f __attribute__((ext_vector_type(4)));
typedef float v8f __attribute__((ext_vector_type(8)));

// LDS (floats): xs[8192] ys[8192] zs[8192] sq[8192] at[8192] bins[512*33]
#define SMEM_FLOATS (5 * N_PTS + THREADS * 33)         // ~227 KB < 320 KB/WGP

__global__ __launch_bounds__(THREADS)
void pointconv_wmma_kernel(const float* __restrict__ pts,
                           const float* __restrict__ attr,
                           const float* __restrict__ W,
                           const float* __restrict__ bias,
                           float* __restrict__ out) {
  extern __shared__ float smem[];
  float* xs = smem;
  float* ys = xs + N_PTS;
  float* zs = ys + N_PTS;
  float* sq = zs + N_PTS;
  float* at = sq + N_PTS;
  float* bins = at + N_PTS;

  const int t = threadIdx.x;

  // ---- Stage points (SoA) + squared norms + attributes into LDS ----
  for (int j = t; j < N_PTS; j += THREADS) {
    float x = pts[3 * j + 0];
    float y = pts[3 * j + 1];
    float z = pts[3 * j + 2];
    xs[j] = x; ys[j] = y; zs[j] = z;
    sq[j] = x * x + y * y + z * z;
    at[j] = attr[j];
  }

  // ---- Zero this lane's private bins (33-dword stride staggers banks) ----
  float* mybin = bins + t * 33;
#pragma unroll
  for (int k = 0; k < 33; ++k) mybin[k] = 0.0f;
  __syncthreads();

  const int lane = t & 31;
  const int wave = t >> 5;
  const int n    = lane & 15;   // column (i) index within tile
  const int hi   = lane >> 4;   // half-wave selector
  const int ibase = (blockIdx.x * WAVES_PER_BLOCK + wave) * TILE;
  const int ipt   = ibase + n;

  const float xi  = xs[ipt];
  const float yi  = ys[ipt];
  const float zi  = zs[ipt];
  const float sqi = sq[ipt];

  // ---- B fragment (4x16, constant over the j loop) ----
  // lanes 0-15: K=0 -> -2x_i, K=1 -> -2y_i ; lanes 16-31: K=2 -> -2z_i, K=3 -> 1
  v2f bfrag;
  bfrag.x = hi ? (-2.0f * zi) : (-2.0f * xi);
  bfrag.y = hi ? 1.0f         : (-2.0f * yi);

  // ---- A fragment per-lane LDS bases (16x4: row m = lane%16) ----
  // lanes 0-15 load (x_j, y_j), lanes 16-31 load (z_j, sq_j)
  const float* a0base = hi ? (zs + n) : (xs + n);
  const float* a1base = hi ? (sq + n) : (ys + n);

  const int rbase = hi * 8; // this half-wave's row (j) group within the tile
  const float thresh = 0.01f - sqi; // d[v] <= thresh  <=>  d2 <= 0.01

#pragma unroll 2
  for (int jb = 0; jb < N_PTS; jb += TILE) {
    // A fragment for this j tile
    v2f afrag;
    afrag.x = a0base[jb];
    afrag.y = a1base[jb];

    v8f c = {};
    // D[j_row, i_col] = sq_j - 2 * dot(p_i, p_j)
    v8f d = __builtin_amdgcn_wmma_f32_16x16x4_f32(
        false, afrag, false, bfrag, (short)0, c, false, false);

    // Fast reject: min over this lane's 8 pairs; tiles skip via execz
    float m01 = fminf(d[0], d[1]);
    float m23 = fminf(d[2], d[3]);
    float m45 = fminf(d[4], d[5]);
    float m67 = fminf(d[6], d[7]);
    float dmin = fminf(fminf(m01, m23), fminf(m45, m67));

    if (dmin <= thresh) {
      // Row (j) coordinates + attributes for this lane's 8 rows (broadcast b128)
      const int jr = jb + rbase;
      float xr[8], yr[8], zr[8], ar[8];
      *(v4f*)(xr)     = *(const v4f*)(xs + jr);
      *(v4f*)(xr + 4) = *(const v4f*)(xs + jr + 4);
      *(v4f*)(yr)     = *(const v4f*)(ys + jr);
      *(v4f*)(yr + 4) = *(const v4f*)(ys + jr + 4);
      *(v4f*)(zr)     = *(const v4f*)(zs + jr);
      *(v4f*)(zr + 4) = *(const v4f*)(zs + jr + 4);
      *(v4f*)(ar)     = *(const v4f*)(at + jr);
      *(v4f*)(ar + 4) = *(const v4f*)(at + jr + 4);

#pragma unroll
      for (int v = 0; v < 8; ++v) {
        float d2 = d[v] + sqi; // full squared distance (ref formula)
        if (d2 <= 0.01f) {     // within radius (negatives from rounding included)
          int cell = (d2 >= 0.0025f ? 8 : 0)   // radial shell (KERNEL_SIZE=2)
                   | (xr[v] > xi ? 4 : 0)      // octant bits: p_j > p_i
                   | (yr[v] > yi ? 2 : 0)
                   | (zr[v] > zi ? 1 : 0);
          atomicAdd(mybin + 2 * cell, ar[v]);     // ds_add_f32, private bins
          atomicAdd(mybin + 2 * cell + 1, 1.0f);  // count
        }
      }
    }
  }
  __syncthreads();

  // ---- Finalize: lanes 0-15 merge with partner lane (+16), mean-pool, contract ----
  if (hi == 0) {
    const float* b0 = bins + t * 33;
    const float* b1 = bins + (t + 16) * 33;
    float mean[16];
#pragma unroll
    for (int k = 0; k < 16; ++k) {
      float s = b0[2 * k] + b1[2 * k];
      float c = b0[2 * k + 1] + b1[2 * k + 1];
      mean[k] = s / fmaxf(c, 1.0f);
    }
#pragma unroll
    for (int o = 0; o < 9; ++o) {
      float acc = bias[o];
#pragma unroll
      for (int k = 0; k < 16; ++k) acc += mean[k] * W[o * 16 + k];
      out[ipt * 9 + o] = acc;
    }
  }
}

extern "C" void kernel_launch(void* const* d_in, const int* in_sizes, int n_in,
                              void* d_out, int out_size, void* d_ws, size_t ws_size,
                              hipStream_t stream) {
  const float* pts  = (const float*)d_in[0]; // (8192, 3)
  const float* attr = (const float*)d_in[1]; // (8192, 1)
  const float* W    = (const float*)d_in[2]; // (9, 1, 16)
  const float* bias = (const float*)d_in[3]; // (9,)
  float* out = (float*)d_out;                // (8192, 9)

  size_t smem_bytes = SMEM_FLOATS * sizeof(float);
  pointconv_wmma_kernel<<<NBLOCKS, THREADS, smem_bytes, stream>>>(pts, attr, W, bias, out);
}